// point_wise_mse_batchweighted_53721450938381
// MI455X (gfx1250) — compile-verified
//
#include <hip/hip_runtime.h>

#define NG 978
#define TK 50
#define PER_LANE 31   // 31 * 32 lanes = 992 >= 978
#define ALPHA 3.85f

typedef float v2f __attribute__((ext_vector_type(2)));
typedef float v8f __attribute__((ext_vector_type(8)));

// ---------- wave32 helpers ----------
__device__ __forceinline__ int wave_reduce_add_i32(int v) {
#pragma unroll
  for (int off = 16; off > 0; off >>= 1) v += __shfl_xor(v, off, 32);
  return v;
}

__device__ __forceinline__ float wave_reduce_add_f32(float v) {
#pragma unroll
  for (int off = 16; off > 0; off >>= 1) v += __shfl_xor(v, off, 32);
  return v;
}

__device__ __forceinline__ int wave_excl_scan_i32(int v, int lane) {
  int x = v;
#pragma unroll
  for (int off = 1; off < 32; off <<= 1) {
    int y = __shfl_up(x, off, 32);
    if (lane >= off) x += y;
  }
  return x - v;
}

// monotone mapping: larger float <-> larger uint
__device__ __forceinline__ unsigned order_key(float f) {
  unsigned u = __float_as_uint(f);
  return (u & 0x80000000u) ? ~u : (u | 0x80000000u);
}

// kth-largest key among all valid keys held by the wave (MSB-first radix select).
// Invalid slots are keyed 0 and can never match a prefix with a set bit.
__device__ __forceinline__ unsigned radix_select_kth(const unsigned* keys, int k) {
  unsigned p = 0;
  int r = k;
  for (int bit = 31; bit >= 0; --bit) {
    unsigned want = (p >> bit) | 1u;
    int c = 0;
#pragma unroll
    for (int j = 0; j < PER_LANE; ++j) c += ((keys[j] >> bit) == want) ? 1 : 0;
    c = wave_reduce_add_i32(c);
    if (c >= r) p |= (1u << bit);
    else r -= c;
  }
  return p;  // value of the kth largest key
}

// Histogram the top-K elements (by key order) of this wave's row into cnt[gene].
__device__ __forceinline__ void hist_topk(const unsigned* keys, int lane,
                                          unsigned* __restrict__ cnt) {
  const unsigned T = radix_select_kth(keys, TK);
  int gcount = 0, eq = 0;
#pragma unroll
  for (int j = 0; j < PER_LANE; ++j) {
    if (keys[j] > T) { atomicAdd(&cnt[lane + 32 * j], 1u); gcount++; }
    else if (keys[j] == T) eq++;
  }
  gcount = wave_reduce_add_i32(gcount);
  const int need = TK - gcount;            // how many elements equal to T to include
  const int pre = wave_excl_scan_i32(eq, lane);
  int taken = 0;
#pragma unroll
  for (int j = 0; j < PER_LANE; ++j) {
    if (keys[j] == T) {
      if (pre + taken < need) atomicAdd(&cnt[lane + 32 * j], 1u);
      taken++;
    }
  }
}

// ---------- kernel 0: zero histograms ----------
__global__ void zero_u32_kernel(unsigned* __restrict__ p, int n) {
  int i = blockIdx.x * blockDim.x + threadIdx.x;
  if (i < n) p[i] = 0u;
}

// ---------- kernel 1: per-row top/bottom-50 histogram (one wave32 per row) ----------
__global__ void topk_hist_kernel(const float* __restrict__ tru,
                                 unsigned* __restrict__ cntTop,
                                 unsigned* __restrict__ cntBot, int B) {
  const int gid  = blockIdx.x * blockDim.x + threadIdx.x;
  const int wave = gid >> 5;
  const int lane = gid & 31;
  if (wave >= B) return;
  const float* row = tru + (size_t)wave * NG;
  unsigned kt[PER_LANE], kb[PER_LANE];
#pragma unroll
  for (int j = 0; j < PER_LANE; ++j) {
    const int g = lane + 32 * j;
    if (g < NG) {
      const unsigned u = order_key(row[g]);
      kt[j] = u;
      kb[j] = ~u;            // bottom-k == top-k of inverted order
    } else {
      kt[j] = 0u;
      kb[j] = 0u;
    }
  }
  hist_topk(kt, lane, cntTop);
  hist_topk(kb, lane, cntBot);
}

// ---------- kernel 2: counts -> combined weight per gene ----------
__global__ void coef_kernel(const unsigned* __restrict__ cntTop,
                            const unsigned* __restrict__ cntBot,
                            float* __restrict__ wt) {
  const int g = blockIdx.x * blockDim.x + threadIdx.x;
  if (g >= NG) return;
  float a = 0.f, b = 0.f;
  const unsigned ct = cntTop[g], cb = cntBot[g];
  if (ct > 0u) a = ALPHA / (1.f + expf((float)ct));  // expf -> inf => 0, as in numpy
  if (cb > 0u) b = ALPHA / (1.f + expf((float)cb));
  wt[g] = a + b;
}

// ---------- kernel 3: WMMA f32 weighted row-sum (one wave per 16-row tile) ----------
// D = A(16x4 se-tile) x B(4x16 broadcast weights) + C, accumulated over genes.
// A layout: lanes 0-15 hold M=lane, K={0,1}; lanes 16-31 hold M=lane-16, K={2,3}.
// B layout: lanes 0-15 hold K rows 0-1 (VGPR0/1); lanes 16-31 hold K rows 2-3.
// Main loop is guard-free (976 = 4*244 genes); the last 2 genes are a scalar tail.
__global__ void wmma_wsum_kernel(const float* __restrict__ pred,
                                 const float* __restrict__ tru,
                                 const float* __restrict__ wt,
                                 float* __restrict__ partials,
                                 int B, int nTiles) {
  const int gid  = blockIdx.x * blockDim.x + threadIdx.x;
  const int wave = gid >> 5;
  const int lane = gid & 31;
  if (wave >= nTiles) return;
  const int m     = lane & 15;   // row within 16-row tile
  const int khalf = lane >> 4;   // 0 -> K{0,1}, 1 -> K{2,3}
  int row = wave * 16 + m;
  const float mask = (row < B) ? 1.f : 0.f;  // branchless tail-tile guard
  if (row >= B) row = 0;                     // keep addresses in-bounds
  // hoist the lane-half K offset into the base pointers; all loop offsets are
  // then uniform immediates and every pair load is 8-byte aligned (978*4 % 8 == 0).
  const float* pbase = pred + (size_t)row * NG + 2 * khalf;
  const float* tbase = tru  + (size_t)row * NG + 2 * khalf;
  const float* wbase = wt + 2 * khalf;

  v8f acc = {};
  constexpr int GMAIN = (NG / 4) * 4;  // 976
#pragma unroll 4
  for (int g = 0; g < GMAIN; g += 4) {
    const v2f p2 = *(const v2f*)(pbase + g);
    const v2f t2 = *(const v2f*)(tbase + g);
    const v2f w2 = *(const v2f*)(wbase + g);
    v2f d = p2 - t2;
    v2f A = (d * d) * mask;
    // 8 args: (neg_a, A, neg_b, B, c_mod, C, reuse_a, reuse_b)
    acc = __builtin_amdgcn_wmma_f32_16x16x4_f32(false, A, false, w2,
                                                (short)0, acc, false, false);
  }
  // D[m][n] is the weighted row-sum of row m, identical for every n.
  // VGPR r holds (M=r, N=lane) on lanes 0-15 and (M=r+8, N=lane-16) on lanes 16-31.
  float s = acc[0] + acc[1] + acc[2] + acc[3] + acc[4] + acc[5] + acc[6] + acc[7];

  // scalar tail: genes [976, 978), one row per lane (lanes 0-15 only)
  float tail = 0.f;
  if (khalf == 0) {
    const float* prow = pbase;  // khalf==0 -> no lane offset
    const float* trow = tbase;
#pragma unroll
    for (int gg = GMAIN; gg < NG; ++gg) {
      const float d = prow[gg] - trow[gg];
      tail += d * d * wt[gg] * mask;
    }
  }
  tail = wave_reduce_add_f32(tail);  // uniform across lanes afterwards

  const float tile = __shfl(s, 0, 32) + __shfl(s, 16, 32) + tail;
  if (lane == 0) partials[wave] = tile;
}

// ---------- kernel 4: deterministic final reduction ----------
__global__ void finalize_kernel(const float* __restrict__ partials, int n,
                                float* __restrict__ out, float scale) {
  __shared__ float sm[256];
  float s = 0.f;
  for (int i = threadIdx.x; i < n; i += 256) s += partials[i];
  sm[threadIdx.x] = s;
  __syncthreads();
  for (int off = 128; off > 0; off >>= 1) {
    if ((int)threadIdx.x < off) sm[threadIdx.x] += sm[threadIdx.x + off];
    __syncthreads();
  }
  if (threadIdx.x == 0) out[0] = sm[0] * scale;
}

extern "C" void kernel_launch(void* const* d_in, const int* in_sizes, int n_in,
                              void* d_out, int out_size, void* d_ws, size_t ws_size,
                              hipStream_t stream) {
  const float* pred = (const float*)d_in[0];
  const float* tru  = (const float*)d_in[1];
  const int B = in_sizes[0] / NG;          // 65536

  // workspace layout
  unsigned* cntTop = (unsigned*)d_ws;      // [NG]
  unsigned* cntBot = cntTop + NG;          // [NG]
  float*    wt     = (float*)(cntBot + NG);// [NG]
  float*    parts  = wt + NG;              // [nTiles]
  const int nTiles = (B + 15) / 16;

  zero_u32_kernel<<<(2 * NG + 255) / 256, 256, 0, stream>>>(cntTop, 2 * NG);
  topk_hist_kernel<<<(B * 32 + 255) / 256, 256, 0, stream>>>(tru, cntTop, cntBot, B);
  coef_kernel<<<(NG + 255) / 256, 256, 0, stream>>>(cntTop, cntBot, wt);
  wmma_wsum_kernel<<<(nTiles * 32 + 255) / 256, 256, 0, stream>>>(pred, tru, wt, parts,
                                                                  B, nTiles);
  finalize_kernel<<<1, 256, 0, stream>>>(parts, nTiles, (float*)d_out,
                                         0.5f / ((float)B * (float)NG));
}